// LSTMModel_37168646980231
// MI455X (gfx1250) — compile-verified
//
#include <hip/hip_runtime.h>
#include <math.h>

// LSTM: B=4096, T=512, IN=3, H=50 (4H=200 gates), then FC [1,H].
// One wave32 per block owns 16 batch rows; recurrent GEMM done with
// V_WMMA_F32_16X16X4_F32 (fp32 matrix path, matches reference precision).

#define B_SZ   4096
#define T_SZ   512
#define IN_SZ  3
#define H_SZ   50
#define G4     200      // 4*H
#define KP     52       // K padded to multiple of 4
#define K2     26       // KP/2 (float2 pairs)
#define NT     13       // gate tiles of 16 -> 208
#define NP     208
#define ROWS   16       // batch rows per block (one wave32)

typedef float v2f __attribute__((ext_vector_type(2)));
typedef float v8f __attribute__((ext_vector_type(8)));

__global__ __launch_bounds__(32)
void lstm_scan_wmma_f32(const float* __restrict__ x,     // [B,T,3]
                        const float* __restrict__ W_ih,  // [200,3]
                        const float* __restrict__ W_hh,  // [200,50]
                        const float* __restrict__ b_ih,  // [200]
                        const float* __restrict__ b_hh,  // [200]
                        const float* __restrict__ W_fc,  // [1,50]
                        const float* __restrict__ b_fc,  // [1]
                        float* __restrict__ out)         // [B,1]
{
    // W_hh^T stored as consecutive-K float2 pairs: Wt2[k2][g] = (W_hh[g][2k2], W_hh[g][2k2+1])
    __shared__ v2f    Wt2[K2][NP];
    __shared__ float4 Wih4[G4];          // (w0, w1, w2, b_ih+b_hh)
    __shared__ float  z_lds[ROWS][NP];   // pre-activation gates for this step
    __shared__ float  h_lds[ROWS][KP];   // hidden state, zero-padded cols 50..51
    __shared__ float  x_lds[ROWS][4];    // x[b, t, 0..2] staged per step

    const int lane = threadIdx.x;        // 0..31 (wave32)
    const int half = lane >> 4;          // 0: lanes 0-15, 1: lanes 16-31
    const int lm   = lane & 15;
    const int rowBase = blockIdx.x * ROWS;

    // ---- stage weights into LDS (zero-padded) ----
    for (int idx = lane; idx < K2 * NP; idx += 32) {
        int k2 = idx / NP, g = idx % NP;
        int k0 = 2 * k2;
        v2f v;
        v.x = (g < G4 && k0     < H_SZ) ? W_hh[g * H_SZ + k0    ] : 0.f;
        v.y = (g < G4 && k0 + 1 < H_SZ) ? W_hh[g * H_SZ + k0 + 1] : 0.f;
        Wt2[k2][g] = v;
    }
    for (int g = lane; g < G4; g += 32) {
        float4 w;
        w.x = W_ih[g * 3 + 0];
        w.y = W_ih[g * 3 + 1];
        w.z = W_ih[g * 3 + 2];
        w.w = b_ih[g] + b_hh[g];
        Wih4[g] = w;
    }
    for (int idx = lane; idx < ROWS * KP; idx += 32)
        (&h_lds[0][0])[idx] = 0.f;       // h0 = 0 (padding stays 0 forever)

    float c_reg[25];                      // c state: 25 (row,j) elements per lane
    #pragma unroll
    for (int i = 0; i < 25; ++i) c_reg[i] = 0.f;

    __syncthreads();

    for (int t = 0; t < T_SZ; ++t) {
        // stage x[rowBase+r, t, 0..2]
        if (lane < ROWS) {
            const float* xp = x + ((size_t)(rowBase + lane) * T_SZ + t) * IN_SZ;
            x_lds[lane][0] = xp[0];
            x_lds[lane][1] = xp[1];
            x_lds[lane][2] = xp[2];
        }

        // ---- A fragments: h (16x52), ISA layout: lanes<16 hold K%4 in {0,1},
        //      lanes>=16 hold K%4 in {2,3}, row M = lane%16 ----
        v2f a[NT];
        #pragma unroll
        for (int s = 0; s < NT; ++s)
            a[s] = *(const v2f*)&h_lds[lm][4 * s + 2 * half];

        // ---- z = h @ W_hh^T via 13x13 WMMA f32 16x16x4 ----
        for (int tt = 0; tt < NT; ++tt) {
            v8f acc = {0.f, 0.f, 0.f, 0.f, 0.f, 0.f, 0.f, 0.f};
            #pragma unroll
            for (int s = 0; s < NT; ++s) {
                v2f b = Wt2[2 * s + half][16 * tt + lm];  // B frag: K pair, N = lane%16
                acc = __builtin_amdgcn_wmma_f32_16x16x4_f32(
                    false, a[s], false, b, (short)0, acc, false, false);
            }
            // D layout: VGPR r -> row M = r + 8*half, col N = lane%16
            #pragma unroll
            for (int r = 0; r < 8; ++r)
                z_lds[r + 8 * half][16 * tt + lm] = acc[r];
        }
        __syncthreads();

        // ---- gate phase: 800 elements = 25 per lane, c kept in registers ----
        #pragma unroll
        for (int it = 0; it < 25; ++it) {
            int e   = lane + 32 * it;
            int row = e / H_SZ;
            int j   = e % H_SZ;
            float x0 = x_lds[row][0], x1 = x_lds[row][1], x2 = x_lds[row][2];
            float4 wi = Wih4[j];
            float4 wf = Wih4[j + 50];
            float4 wg = Wih4[j + 100];
            float4 wo = Wih4[j + 150];
            float pi = z_lds[row][j      ] + wi.w + x0 * wi.x + x1 * wi.y + x2 * wi.z;
            float pf = z_lds[row][j +  50] + wf.w + x0 * wf.x + x1 * wf.y + x2 * wf.z;
            float pg = z_lds[row][j + 100] + wg.w + x0 * wg.x + x1 * wg.y + x2 * wg.z;
            float po = z_lds[row][j + 150] + wo.w + x0 * wo.x + x1 * wo.y + x2 * wo.z;
            float gi = 1.f / (1.f + __expf(-pi));
            float gf = 1.f / (1.f + __expf(-pf));
            float gg = tanhf(pg);
            float go = 1.f / (1.f + __expf(-po));
            float c  = gf * c_reg[it] + gi * gg;
            c_reg[it] = c;
            h_lds[row][j] = go * tanhf(c);
        }
        __syncthreads();
    }

    // ---- final FC: out[b] = h_T[b] . W_fc + b_fc ----
    if (lane < ROWS) {
        float acc = b_fc[0];
        #pragma unroll
        for (int j = 0; j < H_SZ; ++j)
            acc += h_lds[lane][j] * W_fc[j];
        out[rowBase + lane] = acc;
    }
}

extern "C" void kernel_launch(void* const* d_in, const int* in_sizes, int n_in,
                              void* d_out, int out_size, void* d_ws, size_t ws_size,
                              hipStream_t stream) {
    (void)in_sizes; (void)n_in; (void)d_ws; (void)ws_size; (void)out_size;
    const float* x    = (const float*)d_in[0];
    const float* W_ih = (const float*)d_in[1];
    const float* W_hh = (const float*)d_in[2];
    const float* b_ih = (const float*)d_in[3];
    const float* b_hh = (const float*)d_in[4];
    const float* W_fc = (const float*)d_in[5];
    const float* b_fc = (const float*)d_in[6];
    float* out = (float*)d_out;

    dim3 grid(B_SZ / ROWS);   // 256 single-wave workgroups
    dim3 block(32);
    lstm_scan_wmma_f32<<<grid, block, 0, stream>>>(x, W_ih, W_hh, b_ih, b_hh,
                                                   W_fc, b_fc, out);
}